// vertexAdd_29901562315085
// MI455X (gfx1250) — compile-verified
//
#include <hip/hip_runtime.h>
#include <hip/hip_bf16.h>

// Problem constants (from reference): B=32, V=96, F=512, D=3, E=2000
#define NB   32
#define NV   96
#define NF   512
#define ND   3
#define NE   2000
#define EMAX 2048
#define VF   2096   // NV + NE
#define NP   4560   // NV*(NV-1)/2

typedef float v2f __attribute__((ext_vector_type(2)));
typedef float v8f __attribute__((ext_vector_type(8)));

// ---------------------------------------------------------------------------
// Kernel 1: per-batch edge enumeration in triu order (one wave32 per batch).
// Ballot + popc prefix gives each set edge its rank k; edge k -> new vertex
// 96+k. Edge lists + counts land in workspace.
// ---------------------------------------------------------------------------
__global__ __launch_bounds__(32) void build_edges(const float* __restrict__ A,
                                                  int* __restrict__ ei,
                                                  int* __restrict__ ej,
                                                  int* __restrict__ nE) {
    const int b    = blockIdx.x;
    const int lane = threadIdx.x;
    const float* Ab = A + (size_t)b * NV * NV;
    int rank = 0;
    for (int i = 0; i < NV - 1; ++i) {
        const int len = (NV - 1) - i;            // j = i+1 .. 95
        for (int base = 0; base < len; base += 32) {
            const int jo = base + lane;
            bool on = false;
            int  j  = 0;
            if (jo < len) {
                j  = i + 1 + jo;
                on = (Ab[i * NV + j] != 0.0f);
            }
            unsigned m      = (unsigned)__ballot(on);   // wave32: low 32 bits
            unsigned before = m & ((lane == 31) ? 0x7FFFFFFFu : ((1u << lane) - 1u));
            int      k      = rank + __popc(before);
            if (on && k < EMAX) {
                ei[b * EMAX + k] = i;
                ej[b * EMAX + k] = j;
            }
            rank += __popc(m);
        }
    }
    if (lane == 0) nE[b] = (rank < NE) ? rank : NE;
}

// ---------------------------------------------------------------------------
// Kernel 2: B128 zero-fill of the A_new region (562 MB -> the real cost).
// ---------------------------------------------------------------------------
__global__ __launch_bounds__(256) void zeroA(float4* __restrict__ p, size_t n4) {
    size_t idx = (size_t)blockIdx.x * blockDim.x + threadIdx.x;
    if (idx < n4) p[idx] = make_float4(0.f, 0.f, 0.f, 0.f);
}

// ---------------------------------------------------------------------------
// Kernel 3: x_new = S @ x_prev per batch via V_WMMA_F32_16X16X4_F32.
// S rows 0..95 = identity (w=1.0); row 96+k has 0.5 at edge endpoints.
// Exact fp32 arithmetic (weights and zero-products are exact).
// One wave per 16(M) x 16(N) tile; K-loop over 96 in steps of 4.
// A-frag layout (16x4 f32): lanes 0-15 M=0..15 hold K=kk,kk+1; lanes 16-31
// hold K=kk+2,kk+3. B-frag (4x16): mirrored with N across lanes.
// ---------------------------------------------------------------------------
__global__ __launch_bounds__(256) void x_wmma(const float* __restrict__ x_prev,
                                              const int* __restrict__ ei,
                                              const int* __restrict__ ej,
                                              const int* __restrict__ nE,
                                              float* __restrict__ out_x) {
    const int lane  = threadIdx.x;          // 0..31
    const int wid   = threadIdx.y;          // 0..7
    const int mtile = blockIdx.x;           // 0..130  (131*16 = 2096)
    const int ntile = blockIdx.y * 8 + wid; // 0..31   (32*16  = 512)
    const int b     = blockIdx.z;
    const int half  = lane >> 4;
    const int lm    = lane & 15;

    // This lane's A-frag row of S
    const int r  = mtile * 16 + lm;
    int   s0 = -1, s1 = -1;
    float w  = 0.0f;
    if (r < NV) {
        s0 = r; w = 1.0f;                   // identity row
    } else {
        const int k = r - NV;
        if (k < nE[b]) {
            s0 = ei[b * EMAX + k];
            s1 = ej[b * EMAX + k];
            w  = 0.5f;                      // averaging row
        }
    }

    const float* xb   = x_prev + (size_t)b * NV * NF;
    const int    ncol = ntile * 16 + lm;

    v8f c = {};
    #pragma unroll
    for (int kk = 0; kk < NV; kk += 4) {
        const int k0 = kk + 2 * half;       // this lane's first K column
        v2f a, bf;
        a.x = ((s0 == k0)     ? w : 0.0f) + ((s1 == k0)     ? w : 0.0f);
        a.y = ((s0 == k0 + 1) ? w : 0.0f) + ((s1 == k0 + 1) ? w : 0.0f);
        bf.x = xb[(size_t)k0 * NF + ncol];
        bf.y = xb[(size_t)(k0 + 1) * NF + ncol];
        c = __builtin_amdgcn_wmma_f32_16x16x4_f32(
                /*neg_a=*/false, a, /*neg_b=*/false, bf,
                /*c_mod=*/(short)0, c, /*reuse_a=*/false, /*reuse_b=*/false);
    }

    // C/D layout: VGPR v -> M = v + 8*half, N = lm
    float* ox = out_x + (size_t)b * VF * NF;
    #pragma unroll
    for (int v = 0; v < 8; ++v) {
        const int row = mtile * 16 + v + 8 * half;
        ox[(size_t)row * NF + ncol] = c[v];
    }
}

// ---------------------------------------------------------------------------
// Kernel 4: c_new (D=3) — copy rows 0..95, average rows 96+k, zero otherwise.
// ---------------------------------------------------------------------------
__global__ __launch_bounds__(256) void c_kernel(const float* __restrict__ c_prev,
                                                const int* __restrict__ ei,
                                                const int* __restrict__ ej,
                                                const int* __restrict__ nE,
                                                float* __restrict__ out_c) {
    const int t = blockIdx.x * blockDim.x + threadIdx.x;
    if (t >= NB * VF) return;
    const int b = t / VF;
    const int r = t % VF;
    float v0 = 0.f, v1 = 0.f, v2 = 0.f;
    if (r < NV) {
        const float* s = c_prev + ((size_t)b * NV + r) * ND;
        v0 = s[0]; v1 = s[1]; v2 = s[2];
    } else {
        const int k = r - NV;
        if (k < nE[b]) {
            const int i = ei[b * EMAX + k];
            const int j = ej[b * EMAX + k];
            const float* si = c_prev + ((size_t)b * NV + i) * ND;
            const float* sj = c_prev + ((size_t)b * NV + j) * ND;
            v0 = (si[0] + sj[0]) * 0.5f;
            v1 = (si[1] + sj[1]) * 0.5f;
            v2 = (si[2] + sj[2]) * 0.5f;
        }
    }
    float* d = out_c + ((size_t)b * VF + r) * ND;
    d[0] = v0; d[1] = v1; d[2] = v2;
}

// ---------------------------------------------------------------------------
// Kernel 5: scatter the 4 ones per edge into A_new. Targets are provably
// distinct (v unique per edge, i != j, i,j < 96 <= v) -> plain stores.
// ---------------------------------------------------------------------------
__global__ __launch_bounds__(256) void scatterA(const int* __restrict__ ei,
                                                const int* __restrict__ ej,
                                                const int* __restrict__ nE,
                                                float* __restrict__ Aout) {
    const int t = blockIdx.x * blockDim.x + threadIdx.x;
    const int b = t >> 11;          // EMAX = 2048
    const int k = t & (EMAX - 1);
    if (b >= NB) return;
    if (k >= nE[b]) return;
    const int i = ei[b * EMAX + k];
    const int j = ej[b * EMAX + k];
    const int v = NV + k;
    float* Ab = Aout + (size_t)b * VF * VF;
    Ab[(size_t)i * VF + v] = 1.0f;
    Ab[(size_t)v * VF + i] = 1.0f;
    Ab[(size_t)v * VF + j] = 1.0f;
    Ab[(size_t)j * VF + v] = 1.0f;
}

// ---------------------------------------------------------------------------
extern "C" void kernel_launch(void* const* d_in, const int* in_sizes, int n_in,
                              void* d_out, int out_size, void* d_ws, size_t ws_size,
                              hipStream_t stream) {
    (void)in_sizes; (void)n_in; (void)out_size; (void)ws_size;

    const float* x_prev = (const float*)d_in[0];   // (32, 96, 512)
    const float* c_prev = (const float*)d_in[1];   // (32, 96, 3)
    const float* A      = (const float*)d_in[2];   // (32, 96, 96)

    float* out = (float*)d_out;
    const size_t SX = (size_t)NB * VF * NF;        // x_new elements
    const size_t SC = (size_t)NB * VF * ND;        // c_new elements
    const size_t SA = (size_t)NB * VF * VF;        // A_new elements
    float* out_x = out;
    float* out_c = out + SX;
    float* out_A = out + SX + SC;

    // Workspace: edge lists + counts (~524 KB)
    int* ei = (int*)d_ws;
    int* ej = ei + NB * EMAX;
    int* nE = ej + NB * EMAX;

    // 1) enumerate edges (one wave32 per batch)
    build_edges<<<NB, 32, 0, stream>>>(A, ei, ej, nE);

    // 2) zero-fill A_new with B128 stores (the bandwidth-dominant step)
    const size_t n4 = SA / 4;
    zeroA<<<(unsigned)((n4 + 255) / 256), 256, 0, stream>>>((float4*)out_A, n4);

    // 3) x_new via WMMA f32 16x16x4 selection-GEMM
    dim3 xg(VF / 16, 4, NB);    // 131 M-tiles, 4*8 N-tiles, 32 batches
    dim3 xb(32, 8);
    x_wmma<<<xg, xb, 0, stream>>>(x_prev, ei, ej, nE, out_x);

    // 4) c_new
    c_kernel<<<(NB * VF + 255) / 256, 256, 0, stream>>>(c_prev, ei, ej, nE, out_c);

    // 5) scatter ones into A_new (after zeroA on the same stream)
    scatterA<<<(NB * EMAX) / 256, 256, 0, stream>>>(ei, ej, nE, out_A);
}